// RelativePositionBias_58823872086115
// MI455X (gfx1250) — compile-verified
//
#include <hip/hip_runtime.h>
#include <cmath>

typedef __attribute__((ext_vector_type(2))) float v2f;
typedef __attribute__((ext_vector_type(4))) float v4f;
typedef __attribute__((ext_vector_type(8))) float v8f;

#define HIDDEN 32
#define HEADS  16

// ---------------------------------------------------------------------------
// Kernel 1: build tabT[HEADS][ntabp], tabT[n][d_idx] = f_n(d_idx - (T-1))
// One wave (32 lanes) per 16-row tile. Uses V_WMMA_F32_16X16X4_F32:
//   A (16x4 f32): lanes 0-15 = M, comp c holds K = 2*half + c
//   B (4x16 f32): lane gives N = lane&15, comp c holds K = 2*half + c
//   C/D (16x16 f32, 8 VGPRs): N = lane&15, M = v + 8*half
// ---------------------------------------------------------------------------
__global__ __launch_bounds__(32)
void rpb_build_table(const float* __restrict__ w1, const float* __restrict__ b1,
                     const float* __restrict__ w2, const float* __restrict__ b2,
                     float* __restrict__ tab, int T, int ntabp)
{
    const int lane = threadIdx.x;      // 0..31
    const int tile = blockIdx.x;       // 16 d-values per tile
    const int m    = lane & 15;        // M for A; N for B/C/D
    const int half = lane >> 4;

    // signed log1p transform of relative distance for this A-row
    const int   idx = tile * 16 + m;
    const float fd  = (float)(idx - (T - 1));
    const float r   = copysignf(log1pf(fabsf(fd)), fd);   // sign(d)*log1p(|d|)

    // seed accumulator with b2[n] (n = lane&15 in the C/D layout)
    v8f acc;
    const float bias2 = b2[m];
#pragma unroll
    for (int v = 0; v < 8; ++v) acc[v] = bias2;

#pragma unroll
    for (int kt = 0; kt < HIDDEN / 4; ++kt) {
        v2f a, b;
#pragma unroll
        for (int c = 0; c < 2; ++c) {
            const int k = kt * 4 + half * 2 + c;          // global K in [0,32)
            const float hv = fmaf(r, w1[k], b1[k]);       // rel*w1 + b1
            a[c] = hv > 0.0f ? hv : 0.0f;                 // relu
            b[c] = w2[k * HEADS + m];                     // w2[k][n]
        }
        // (neg_a, A, neg_b, B, c_mod, C, reuse_a, reuse_b)
        acc = __builtin_amdgcn_wmma_f32_16x16x4_f32(
                  false, a, false, b, (short)0, acc, false, false);
    }

    // scatter D into transposed table: tab[n][tile*16 + M]
#pragma unroll
    for (int v = 0; v < 8; ++v) {
        const int M = v + half * 8;
        tab[(size_t)m * (size_t)ntabp + (size_t)(tile * 16 + M)] = acc[v];
    }
}

// ---------------------------------------------------------------------------
// Kernel 2: out[n][i][j] = tabT[n][i - j + (T-1)]
// grid = (T/1024, T, HEADS); each thread: 4 coalesced cached loads,
// one 16B non-temporal store (pure HBM store stream, ~256 MB).
// ---------------------------------------------------------------------------
__global__ __launch_bounds__(256)
void rpb_broadcast(const float* __restrict__ tab, float* __restrict__ out,
                   int T, int ntabp)
{
    const int n = blockIdx.z;
    const int i = blockIdx.y;
    const int j = (blockIdx.x * blockDim.x + threadIdx.x) * 4;
    if (j >= T) return;

    const float* __restrict__ row = tab + (size_t)n * (size_t)ntabp;
    const int    td0  = i - j + (T - 1);
    const size_t base = ((size_t)n * (size_t)T + (size_t)i) * (size_t)T + (size_t)j;

    if (j + 3 < T) {
        v4f v = { row[td0], row[td0 - 1], row[td0 - 2], row[td0 - 3] };
        __builtin_nontemporal_store(v, (v4f*)(out + base));
    } else {
        for (int q = 0; q < 4 && (j + q) < T; ++q)
            out[base + q] = row[td0 - q];
    }
}

// ---------------------------------------------------------------------------
extern "C" void kernel_launch(void* const* d_in, const int* in_sizes, int n_in,
                              void* d_out, int out_size, void* d_ws, size_t ws_size,
                              hipStream_t stream)
{
    // inputs: [0]=T (unused device scalar), [1]=w1[1,32], [2]=b1[32],
    //         [3]=w2[32,16], [4]=b2[16]
    const float* w1 = (const float*)d_in[1];
    const float* b1 = (const float*)d_in[2];
    const float* w2 = (const float*)d_in[3];
    const float* b2 = (const float*)d_in[4];
    float* out = (float*)d_out;
    float* tab = (float*)d_ws;

    // out is [HEADS, T, T] -> derive T (T is passed only as a device scalar)
    const long long tt = (long long)out_size / HEADS;
    const int T = (int)(sqrt((double)tt) + 0.5);

    const int ntab  = 2 * T - 1;
    const int ntabp = (ntab + 15) & ~15;     // pad to WMMA tile multiple (4096)
    const int tiles = ntabp / 16;

    rpb_build_table<<<dim3(tiles), dim3(32), 0, stream>>>(w1, b1, w2, b2, tab, T, ntabp);

    const int jblocks = (T + 1023) / 1024;   // 256 threads * 4 floats = 1024 j per block
    rpb_broadcast<<<dim3(jblocks, T, HEADS), dim3(256), 0, stream>>>(tab, out, T, ntabp);
}